// Model_56169582297507
// MI455X (gfx1250) — compile-verified
//
#include <hip/hip_runtime.h>
#include <math.h>

// ---------------- problem constants (match reference) ----------------
#define NN        100000      // nodes
#define NE        1000000     // edges
#define F_IN      128
#define F_HID     32
#define HEADS     8
#define HF        (HEADS * F_HID)   // 256
#define NCLS      16
#define NEG_SLOPE 0.2f

typedef __attribute__((ext_vector_type(2))) float v2f;
typedef __attribute__((ext_vector_type(8))) float v8f;

__device__ __forceinline__ void async_copy_b128(unsigned ldsOff, const float* g) {
  // GLOBAL_LOAD_ASYNC_TO_LDS_B128: copies 16 bytes from global memory into LDS at vdst, tracked by ASYNCcnt
  asm volatile("global_load_async_to_lds_b128 %0, %1, off"
               :: "v"(ldsOff), "v"(g) : "memory");
}

// =====================================================================
// GEMM1: h1[N,256] = x[N,128] @ W1[128,256]   (fp32 WMMA 16x16x4)
// block = 256 threads = 8 waves; each wave computes rows [r0, r0+16).
// W1 column tiles (128x16 = 8KB) staged in LDS via async-to-LDS,
// double-buffered against the WMMA loop. A fragments live in registers.
// B fragments for a whole tile are batch-loaded from LDS into registers
// first, so the 32 chained WMMAs issue without per-step dscnt-0 stalls.
// =====================================================================
__global__ __launch_bounds__(256) void gemm1_wmma(
    const float* __restrict__ x, const float* __restrict__ W,
    float* __restrict__ h1) {
  __shared__ __align__(16) float Wt[2][128 * 16];   // 16 KB double buffer
  const int tid  = threadIdx.x;
  const int lane = tid & 31;
  const int wave = tid >> 5;
  const int r0   = blockIdx.x * 128 + wave * 16;
  const int half = lane >> 4;   // 0: K offset {0,1}; 1: K offset {2,3}
  const int l16  = lane & 15;

  const int  row   = r0 + l16;
  const bool rowok = row < NN;
  const int  rsafe = rowok ? row : 0;

  // A fragments: 16x128 tile -> 32 K-steps of float2 per lane
  v2f a[32];
  const float* xr = x + (size_t)rsafe * F_IN + half * 2;
#pragma unroll
  for (int kk = 0; kk < 32; ++kk) {
    float av0 = xr[kk * 4 + 0];
    float av1 = xr[kk * 4 + 1];
    a[kk].x = rowok ? av0 : 0.0f;
    a[kk].y = rowok ? av1 : 0.0f;
  }

  // cooperative async stage of W column tile ct into buffer b:
  // 128 rows x 16 cols = 512 float4s; 256 threads x 2 b128 transfers
  auto issue = [&](int ct, int b) {
#pragma unroll
    for (int i = 0; i < 2; ++i) {
      const int q  = tid + i * 256;        // float4 index in tile
      const int r  = q >> 2;
      const int c4 = (q & 3) << 2;
      const unsigned ldsOff = (unsigned)(uintptr_t)&Wt[b][r * 16 + c4];
      async_copy_b128(ldsOff, W + (size_t)r * HF + ct * 16 + c4);
    }
  };

  issue(0, 0);                              // prefetch first tile

  for (int ct = 0; ct < 16; ++ct) {         // 16 column tiles of 16
    const int cur = ct & 1;
    if (ct < 15) {
      issue(ct + 1, cur ^ 1);               // prefetch next tile
      asm volatile("s_wait_asynccnt 0x2" ::: "memory");  // tile ct complete
    } else {
      asm volatile("s_wait_asynccnt 0x0" ::: "memory");
    }
    __syncthreads();                        // tile ct visible to all waves

    const float* Wl = &Wt[cur][l16];
    // batch-load all B fragments for this tile (ds_load clause),
    // then run the chained WMMA sequence without load-use stalls
    v2f bfr[32];
#pragma unroll
    for (int kk = 0; kk < 32; ++kk) {
      const int krow = kk * 4 + half * 2;
      bfr[kk].x = Wl[krow * 16];
      bfr[kk].y = Wl[krow * 16 + 16];
    }
    v8f acc = {0.f, 0.f, 0.f, 0.f, 0.f, 0.f, 0.f, 0.f};
#pragma unroll
    for (int kk = 0; kk < 32; ++kk) {
      acc = __builtin_amdgcn_wmma_f32_16x16x4_f32(
          false, a[kk], false, bfr[kk], (short)0, acc, false, false);
    }
#pragma unroll
    for (int r = 0; r < 8; ++r) {
      const int orow = r0 + r + half * 8;   // C/D layout: lanes16-31 -> M+8
      if (orow < NN) h1[(size_t)orow * HF + ct * 16 + l16] = acc[r];
    }
    __syncthreads();                        // done reading before overwrite
  }
}

// =====================================================================
// GEMM2: h2[N,16] = elu(agg1 + b1)[N,256] @ W2[256,16]
// bias + ELU fused into the A-operand load (agg1 post-activation is
// consumed nowhere else), W2 (16KB) staged fully in LDS via async copy.
// K processed in 4 groups of 16 steps: stage A+B frags, then 16 WMMAs.
// =====================================================================
__global__ __launch_bounds__(256) void gemm2_wmma(
    const float* __restrict__ A, const float* __restrict__ W,
    const float* __restrict__ bias, float* __restrict__ h2) {
  __shared__ __align__(16) float Ws[256 * 16];     // full W2, 16 KB
  const int tid  = threadIdx.x;
  const int lane = tid & 31;
  const int wave = tid >> 5;
  const int r0   = blockIdx.x * 128 + wave * 16;
  const int half = lane >> 4;
  const int l16  = lane & 15;

  // stage W2: 4096 floats = 1024 float4s; 256 threads x 4 b128 transfers
#pragma unroll
  for (int i = 0; i < 4; ++i) {
    const int q = tid + i * 256;            // float4 index
    const unsigned ldsOff = (unsigned)(uintptr_t)&Ws[q * 4];
    async_copy_b128(ldsOff, W + (size_t)q * 4);
  }

  const int  row   = r0 + l16;
  const bool rowok = row < NN;
  const int  rsafe = rowok ? row : 0;
  const float* Ar = A + (size_t)rsafe * HF + half * 2;

  asm volatile("s_wait_asynccnt 0x0" ::: "memory");
  __syncthreads();

  v8f acc = {0.f, 0.f, 0.f, 0.f, 0.f, 0.f, 0.f, 0.f};
  for (int g = 0; g < 4; ++g) {             // K = 256 -> 4 groups x 16 steps
    v2f av[16], bv[16];
#pragma unroll
    for (int j = 0; j < 16; ++j) {
      const int kk = g * 16 + j;
      const int k0 = kk * 4 + half * 2;
      float a0 = Ar[kk * 4 + 0] + bias[k0 + 0];
      float a1 = Ar[kk * 4 + 1] + bias[k0 + 1];
      a0 = a0 > 0.f ? a0 : expm1f(a0);      // fused ELU
      a1 = a1 > 0.f ? a1 : expm1f(a1);
      av[j].x = rowok ? a0 : 0.0f;
      av[j].y = rowok ? a1 : 0.0f;
      bv[j].x = Ws[(k0 + 0) * NCLS + l16];
      bv[j].y = Ws[(k0 + 1) * NCLS + l16];
    }
#pragma unroll
    for (int j = 0; j < 16; ++j) {
      acc = __builtin_amdgcn_wmma_f32_16x16x4_f32(
          false, av[j], false, bv[j], (short)0, acc, false, false);
    }
  }
#pragma unroll
  for (int r = 0; r < 8; ++r) {
    const int orow = r0 + r + half * 8;
    if (orow < NN) h2[(size_t)orow * NCLS + l16] = acc[r];
  }
}

// =====================================================================
// Layer-1 scores + init: s_src/s_dst[n,h] = <h1[n,h,:], a_*[h,:]>,
// m = -inf, denom = 0, agg1 = 0. One thread per (node, head).
// =====================================================================
__global__ void scores_init1(const float* __restrict__ h1,
                             const float* __restrict__ a_src,
                             const float* __restrict__ a_dst,
                             float* __restrict__ s_src, float* __restrict__ s_dst,
                             float* __restrict__ m, float* __restrict__ den,
                             float* __restrict__ agg) {
  const int t = blockIdx.x * blockDim.x + threadIdx.x;
  if (t >= NN * HEADS) return;
  const int n = t >> 3, hh = t & 7;
  const float* hp = h1 + (size_t)n * HF + hh * F_HID;
  const float* as = a_src + hh * F_HID;
  const float* ad = a_dst + hh * F_HID;
  float ss = 0.f, sd = 0.f;
#pragma unroll
  for (int j = 0; j < F_HID; ++j) {
    const float v = hp[j];
    ss += v * as[j];
    sd += v * ad[j];
  }
  s_src[t] = ss;
  s_dst[t] = sd;
  m[t]   = -INFINITY;
  den[t] = 0.f;
  float4* ag = (float4*)(agg + (size_t)n * HF + hh * F_HID);
#pragma unroll
  for (int j = 0; j < F_HID / 4; ++j) ag[j] = make_float4(0.f, 0.f, 0.f, 0.f);
}

__device__ __forceinline__ float leaky(float v) {
  return v > 0.f ? v : NEG_SLOPE * v;
}

// pass 1: segment max over dst
__global__ void edge_max1(const int* __restrict__ ei,
                          const float* __restrict__ s_src,
                          const float* __restrict__ s_dst,
                          float* __restrict__ m) {
  const int t = blockIdx.x * blockDim.x + threadIdx.x;
  if (t >= NE * HEADS) return;
  const int e = t >> 3, hh = t & 7;
  const int s = ei[e], d = ei[NE + e];
  const float v = leaky(s_src[s * HEADS + hh] + s_dst[d * HEADS + hh]);
  atomicMax(&m[d * HEADS + hh], v);      // global_atomic_max_num_f32
}

// pass 2: segment sum of exp(e - m[dst])
__global__ void edge_sum1(const int* __restrict__ ei,
                          const float* __restrict__ s_src,
                          const float* __restrict__ s_dst,
                          const float* __restrict__ m,
                          float* __restrict__ den) {
  const int t = blockIdx.x * blockDim.x + threadIdx.x;
  if (t >= NE * HEADS) return;
  const int e = t >> 3, hh = t & 7;
  const int s = ei[e], d = ei[NE + e];
  const float v = leaky(s_src[s * HEADS + hh] + s_dst[d * HEADS + hh]);
  atomicAdd(&den[d * HEADS + hh], __expf(v - m[d * HEADS + hh]));
}

// pass 3: agg[dst] += alpha * h1[src]   (32 floats per (edge,head))
__global__ void edge_msg1(const int* __restrict__ ei,
                          const float* __restrict__ s_src,
                          const float* __restrict__ s_dst,
                          const float* __restrict__ m,
                          const float* __restrict__ den,
                          const float* __restrict__ h1,
                          float* __restrict__ agg) {
  const int t = blockIdx.x * blockDim.x + threadIdx.x;
  if (t >= NE * HEADS) return;
  const int e = t >> 3, hh = t & 7;
  const int s = ei[e], d = ei[NE + e];
  const float v  = leaky(s_src[s * HEADS + hh] + s_dst[d * HEADS + hh]);
  const float al = __expf(v - m[d * HEADS + hh]) / (den[d * HEADS + hh] + 1e-16f);
  const float4* hp = (const float4*)(h1 + (size_t)s * HF + hh * F_HID);
  float* ag = agg + (size_t)d * HF + hh * F_HID;
#pragma unroll
  for (int j = 0; j < F_HID / 4; ++j) {
    const float4 q = hp[j];
    atomicAdd(ag + j * 4 + 0, q.x * al);
    atomicAdd(ag + j * 4 + 1, q.y * al);
    atomicAdd(ag + j * 4 + 2, q.z * al);
    atomicAdd(ag + j * 4 + 3, q.w * al);
  }
}

// ---------------- layer 2 (H=1, F=16) ----------------
__global__ void scores_init2(const float* __restrict__ h2,
                             const float* __restrict__ a_src,
                             const float* __restrict__ a_dst,
                             float* __restrict__ s_src, float* __restrict__ s_dst,
                             float* __restrict__ m, float* __restrict__ den,
                             float* __restrict__ agg) {
  const int n = blockIdx.x * blockDim.x + threadIdx.x;
  if (n >= NN) return;
  const float* hp = h2 + (size_t)n * NCLS;
  float ss = 0.f, sd = 0.f;
#pragma unroll
  for (int c = 0; c < NCLS; ++c) {
    const float v = hp[c];
    ss += v * a_src[c];
    sd += v * a_dst[c];
  }
  s_src[n] = ss;
  s_dst[n] = sd;
  m[n]   = -INFINITY;
  den[n] = 0.f;
  float4* ag = (float4*)(agg + (size_t)n * NCLS);
#pragma unroll
  for (int j = 0; j < NCLS / 4; ++j) ag[j] = make_float4(0.f, 0.f, 0.f, 0.f);
}

__global__ void edge_max2(const int* __restrict__ ei,
                          const float* __restrict__ s_src,
                          const float* __restrict__ s_dst,
                          float* __restrict__ m) {
  const int e = blockIdx.x * blockDim.x + threadIdx.x;
  if (e >= NE) return;
  const int s = ei[e], d = ei[NE + e];
  atomicMax(&m[d], leaky(s_src[s] + s_dst[d]));
}

__global__ void edge_sum2(const int* __restrict__ ei,
                          const float* __restrict__ s_src,
                          const float* __restrict__ s_dst,
                          const float* __restrict__ m,
                          float* __restrict__ den) {
  const int e = blockIdx.x * blockDim.x + threadIdx.x;
  if (e >= NE) return;
  const int s = ei[e], d = ei[NE + e];
  atomicAdd(&den[d], __expf(leaky(s_src[s] + s_dst[d]) - m[d]));
}

__global__ void edge_msg2(const int* __restrict__ ei,
                          const float* __restrict__ s_src,
                          const float* __restrict__ s_dst,
                          const float* __restrict__ m,
                          const float* __restrict__ den,
                          const float* __restrict__ h2,
                          float* __restrict__ agg) {
  const int e = blockIdx.x * blockDim.x + threadIdx.x;
  if (e >= NE) return;
  const int s = ei[e], d = ei[NE + e];
  const float v  = leaky(s_src[s] + s_dst[d]);
  const float al = __expf(v - m[d]) / (den[d] + 1e-16f);
  const float4* hp = (const float4*)(h2 + (size_t)s * NCLS);
  float* ag = agg + (size_t)d * NCLS;
#pragma unroll
  for (int j = 0; j < NCLS / 4; ++j) {
    const float4 q = hp[j];
    atomicAdd(ag + j * 4 + 0, q.x * al);
    atomicAdd(ag + j * 4 + 1, q.y * al);
    atomicAdd(ag + j * 4 + 2, q.z * al);
    atomicAdd(ag + j * 4 + 3, q.w * al);
  }
}

__global__ void finalize2(const float* __restrict__ agg,
                          const float* __restrict__ b,
                          float* __restrict__ out) {
  const int t = blockIdx.x * blockDim.x + threadIdx.x;
  if (t >= NN * NCLS) return;
  out[t] = agg[t] + b[t & (NCLS - 1)];
}

// =====================================================================
extern "C" void kernel_launch(void* const* d_in, const int* in_sizes, int n_in,
                              void* d_out, int out_size, void* d_ws, size_t ws_size,
                              hipStream_t stream) {
  const float* x      = (const float*)d_in[0];
  const int*   ei     = (const int*)d_in[1];     // [2, NE] flat: src then dst
  const float* W1     = (const float*)d_in[2];
  const float* a_src1 = (const float*)d_in[3];
  const float* a_dst1 = (const float*)d_in[4];
  const float* b1     = (const float*)d_in[5];
  const float* W2     = (const float*)d_in[6];
  const float* a_src2 = (const float*)d_in[7];
  const float* a_dst2 = (const float*)d_in[8];
  const float* b2     = (const float*)d_in[9];
  float* out = (float*)d_out;

  // workspace layout (floats); layer-2 buffers alias the dead h1 region
  float* ws     = (float*)d_ws;
  float* h1     = ws;                               // NN*256
  float* agg1   = ws + (size_t)NN * HF;             // NN*256
  float* s_src1 = ws + (size_t)2 * NN * HF;         // NN*8
  float* s_dst1 = s_src1 + (size_t)NN * HEADS;      // NN*8
  float* m1     = s_dst1 + (size_t)NN * HEADS;      // NN*8
  float* den1   = m1 + (size_t)NN * HEADS;          // NN*8
  float* h2     = h1;                               // NN*16 (aliases h1)
  float* s_src2 = h1 + (size_t)NN * NCLS;           // NN
  float* s_dst2 = s_src2 + NN;
  float* m2     = s_dst2 + NN;
  float* den2   = m2 + NN;
  float* agg2   = den2 + NN;                        // NN*16

  const int T = 256;
  const int gRows = (NN + 127) / 128;               // 8 waves x 16 rows

  // ---- layer 1 ----
  gemm1_wmma<<<gRows, T, 0, stream>>>(x, W1, h1);
  scores_init1<<<(NN * HEADS + T - 1) / T, T, 0, stream>>>(
      h1, a_src1, a_dst1, s_src1, s_dst1, m1, den1, agg1);
  edge_max1<<<(NE * HEADS + T - 1) / T, T, 0, stream>>>(ei, s_src1, s_dst1, m1);
  edge_sum1<<<(NE * HEADS + T - 1) / T, T, 0, stream>>>(ei, s_src1, s_dst1, m1, den1);
  edge_msg1<<<(NE * HEADS + T - 1) / T, T, 0, stream>>>(
      ei, s_src1, s_dst1, m1, den1, h1, agg1);

  // ---- layer 2 (bias + ELU fused into GEMM2 A-operand) ----
  gemm2_wmma<<<gRows, T, 0, stream>>>(agg1, W2, b1, h2);
  scores_init2<<<(NN + T - 1) / T, T, 0, stream>>>(
      h2, a_src2, a_dst2, s_src2, s_dst2, m2, den2, agg2);
  edge_max2<<<(NE + T - 1) / T, T, 0, stream>>>(ei, s_src2, s_dst2, m2);
  edge_sum2<<<(NE + T - 1) / T, T, 0, stream>>>(ei, s_src2, s_dst2, m2, den2);
  edge_msg2<<<(NE + T - 1) / T, T, 0, stream>>>(
      ei, s_src2, s_dst2, m2, den2, h2, agg2);
  finalize2<<<(NN * NCLS + T - 1) / T, T, 0, stream>>>(agg2, b2, out);
}